// GCN_51445118271731
// MI455X (gfx1250) — compile-verified
//
#include <hip/hip_runtime.h>
#include <hip/hip_bf16.h>
#include <math.h>

// ---------------------------------------------------------------------------
// GCN forward for MI455X (gfx1250, wave32).
//  - GEMMs (x@W1, relu(h)@W2) use V_WMMA_F32_16X16X4_F32 (exact fp32 WMMA).
//  - Edge aggregation is L2-resident scatter-add via global_atomic_add_f32.
// ---------------------------------------------------------------------------

typedef __attribute__((ext_vector_type(2))) float v2f;
typedef __attribute__((ext_vector_type(8))) float v8f;

#define F_IN     128
#define HIDDEN   16
#define N_CLASS  10
#define N_GRAPHS 512

// ---------------- degree / norm ----------------

__global__ void init_deg(float* deg, int n) {
    int i = blockIdx.x * blockDim.x + threadIdx.x;
    if (i < n) deg[i] = 1.0f;   // self-loop contributes 1
}

__global__ void deg_edges(const int* __restrict__ dst, float* deg, int E) {
    int e = blockIdx.x * blockDim.x + threadIdx.x;
    if (e < E) atomicAdd(&deg[dst[e]], 1.0f);
}

__global__ void make_dinv(const float* __restrict__ deg, float* dinv, int n) {
    int i = blockIdx.x * blockDim.x + threadIdx.x;
    if (i < n) dinv[i] = rsqrtf(fmaxf(deg[i], 1e-12f));
}

// ---------------- GEMM1: h = x @ W1   (N x 128) @ (128 x 16) ----------------
// One wave32 computes one 16x16 output tile via 32 chained f32 WMMAs.

__global__ __launch_bounds__(256) void gemm1_wmma(const float* __restrict__ x,
                                                  const float* __restrict__ W,
                                                  float* __restrict__ h,
                                                  int n_nodes) {
    __shared__ float sW[F_IN * HIDDEN];           // 8 KB
    for (int i = threadIdx.x; i < F_IN * HIDDEN; i += 256) sW[i] = W[i];
    __syncthreads();

    int wave = threadIdx.x >> 5;
    int lane = threadIdx.x & 31;
    int tile = blockIdx.x * 8 + wave;
    if (tile * 16 >= n_nodes) return;             // wave-uniform: EXEC stays all-1s

    int m  = lane & 15;                           // A row / B col for this lane
    int kb = (lane >> 4) << 1;                    // K sub-offset: 0 or 2

    const float* xr = x + (size_t)(tile * 16 + m) * F_IN;

    v8f c = {};
    #pragma unroll
    for (int kk = 0; kk < F_IN / 4; ++kk) {
        int k0 = kk * 4 + kb;
        v2f a, b;
        a.x = xr[k0];
        a.y = xr[k0 + 1];
        b.x = sW[k0 * HIDDEN + m];
        b.y = sW[(k0 + 1) * HIDDEN + m];
        c = __builtin_amdgcn_wmma_f32_16x16x4_f32(false, a, false, b,
                                                  (short)0, c, false, false);
    }

    int row0 = tile * 16 + ((lane < 16) ? 0 : 8);
    #pragma unroll
    for (int r = 0; r < 8; ++r)
        h[(size_t)(row0 + r) * HIDDEN + m] = c[r];
}

// ---------------- GEMM2: h2 = relu(a1) @ W2   (N x 16) @ (16 x 16) ----------

__global__ __launch_bounds__(256) void gemm2_wmma(const float* __restrict__ a1,
                                                  const float* __restrict__ W,
                                                  float* __restrict__ h,
                                                  int n_nodes) {
    __shared__ float sW[HIDDEN * HIDDEN];         // 1 KB
    if (threadIdx.x < HIDDEN * HIDDEN) sW[threadIdx.x] = W[threadIdx.x];
    __syncthreads();

    int wave = threadIdx.x >> 5;
    int lane = threadIdx.x & 31;
    int tile = blockIdx.x * 8 + wave;
    if (tile * 16 >= n_nodes) return;

    int m  = lane & 15;
    int kb = (lane >> 4) << 1;

    const float* ar = a1 + (size_t)(tile * 16 + m) * HIDDEN;

    v8f c = {};
    #pragma unroll
    for (int kk = 0; kk < HIDDEN / 4; ++kk) {
        int k0 = kk * 4 + kb;
        v2f a, b;
        a.x = fmaxf(ar[k0], 0.0f);                // fused ReLU
        a.y = fmaxf(ar[k0 + 1], 0.0f);
        b.x = sW[k0 * HIDDEN + m];
        b.y = sW[(k0 + 1) * HIDDEN + m];
        c = __builtin_amdgcn_wmma_f32_16x16x4_f32(false, a, false, b,
                                                  (short)0, c, false, false);
    }

    int row0 = tile * 16 + ((lane < 16) ? 0 : 8);
    #pragma unroll
    for (int r = 0; r < 8; ++r)
        h[(size_t)(row0 + r) * HIDDEN + m] = c[r];
}

// -------- self-loop + bias init:  out[n,j] = b[j] + h[n,j] * dinv[n]^2 ------

__global__ void self_init(const float* __restrict__ h,
                          const float* __restrict__ dinv,
                          const float* __restrict__ bias,
                          float* __restrict__ out, int n_nodes) {
    int t = blockIdx.x * blockDim.x + threadIdx.x;
    if (t >= n_nodes * HIDDEN) return;
    int n = t >> 4;
    int j = t & 15;
    float di = dinv[n];
    out[t] = bias[j] + h[t] * di * di;
}

// -------- edge scatter:  out[dst] += h[src] * dinv[src]*dinv[dst] -----------
// 4 threads per edge, float4 gather, 4 f32 atomics each. L2-resident.

__global__ void edge_scatter(const int* __restrict__ src,
                             const int* __restrict__ dst,
                             const float* __restrict__ dinv,
                             const float* __restrict__ h,
                             float* __restrict__ out, int E) {
    int t = blockIdx.x * blockDim.x + threadIdx.x;
    int e = t >> 2;
    if (e >= E) return;
    int s = src[e], d = dst[e];
    float nrm = dinv[s] * dinv[d];
    int c = (t & 3) * 4;
    const float4 v = *(const float4*)(h + (size_t)s * HIDDEN + c);
    float* o = out + (size_t)d * HIDDEN + c;
    atomicAdd(o + 0, v.x * nrm);
    atomicAdd(o + 1, v.y * nrm);
    atomicAdd(o + 2, v.z * nrm);
    atomicAdd(o + 3, v.w * nrm);
}

// ---------------- global mean pool ----------------

__global__ void pool_init(float* sums, float* cnt) {
    int i = blockIdx.x * blockDim.x + threadIdx.x;
    if (i < N_GRAPHS * HIDDEN) sums[i] = 0.0f;
    if (i < N_GRAPHS) cnt[i] = 0.0f;
}

__global__ void pool_scatter(const float* __restrict__ x,
                             const int* __restrict__ batch,
                             float* sums, float* cnt, int n_nodes) {
    int t = blockIdx.x * blockDim.x + threadIdx.x;
    if (t >= n_nodes * HIDDEN) return;
    int n = t >> 4;
    int j = t & 15;
    int g = batch[n];
    atomicAdd(&sums[g * HIDDEN + j], x[t]);
    if (j == 0) atomicAdd(&cnt[g], 1.0f);
}

// ---------------- classifier + log_softmax ----------------

__global__ void finalize(const float* __restrict__ sums,
                         const float* __restrict__ cnt,
                         const float* __restrict__ Wl,
                         const float* __restrict__ bl,
                         float* __restrict__ out) {
    int g = blockIdx.x * blockDim.x + threadIdx.x;
    if (g >= N_GRAPHS) return;
    float inv = 1.0f / fmaxf(cnt[g], 1.0f);
    float mean[HIDDEN];
    #pragma unroll
    for (int j = 0; j < HIDDEN; ++j) mean[j] = sums[g * HIDDEN + j] * inv;
    float lg[N_CLASS];
    #pragma unroll
    for (int c = 0; c < N_CLASS; ++c) {
        float acc = bl[c];
        #pragma unroll
        for (int j = 0; j < HIDDEN; ++j) acc += mean[j] * Wl[j * N_CLASS + c];
        lg[c] = acc;
    }
    float mx = lg[0];
    #pragma unroll
    for (int c = 1; c < N_CLASS; ++c) mx = fmaxf(mx, lg[c]);
    float s = 0.0f;
    #pragma unroll
    for (int c = 0; c < N_CLASS; ++c) s += __expf(lg[c] - mx);
    float lse = __logf(s);
    #pragma unroll
    for (int c = 0; c < N_CLASS; ++c) out[g * N_CLASS + c] = lg[c] - mx - lse;
}

// ---------------------------------------------------------------------------

extern "C" void kernel_launch(void* const* d_in, const int* in_sizes, int n_in,
                              void* d_out, int out_size, void* d_ws, size_t ws_size,
                              hipStream_t stream) {
    const float* x   = (const float*)d_in[0];
    const int*   ei  = (const int*)  d_in[1];
    const int*   bat = (const int*)  d_in[2];
    const float* W1  = (const float*)d_in[3];
    const float* b1  = (const float*)d_in[4];
    const float* W2  = (const float*)d_in[5];
    const float* b2  = (const float*)d_in[6];
    const float* Wl  = (const float*)d_in[7];
    const float* bl  = (const float*)d_in[8];
    float* out = (float*)d_out;

    const int N = in_sizes[0] / F_IN;   // 100000
    const int E = in_sizes[1] / 2;      // 3200000
    const int* src = ei;
    const int* dst = ei + E;

    // workspace layout (floats)
    float* deg  = (float*)d_ws;                 // N
    float* dinv = deg  + N;                     // N
    float* bufA = dinv + N;                     // N*16  (h1, then h2)
    float* bufB = bufA + (size_t)N * HIDDEN;    // N*16  (out1, then out2)
    float* sums = bufB + (size_t)N * HIDDEN;    // 512*16
    float* cnt  = sums + N_GRAPHS * HIDDEN;     // 512

    const int B = 256;
    int gN   = (N + B - 1) / B;
    int gE   = (E + B - 1) / B;
    int gNH  = (N * HIDDEN + B - 1) / B;
    int gE4  = (E * 4 + B - 1) / B;
    int tiles = (N + 15) / 16;
    int gT   = (tiles + 7) / 8;

    // degree & symmetric norm
    init_deg  <<<gN, B, 0, stream>>>(deg, N);
    deg_edges <<<gE, B, 0, stream>>>(dst, deg, E);
    make_dinv <<<gN, B, 0, stream>>>(deg, dinv, N);

    // layer 1
    gemm1_wmma  <<<gT,  B, 0, stream>>>(x, W1, bufA, N);
    self_init   <<<gNH, B, 0, stream>>>(bufA, dinv, b1, bufB, N);
    edge_scatter<<<gE4, B, 0, stream>>>(src, dst, dinv, bufA, bufB, E);

    // layer 2 (ReLU fused into GEMM2 A-load)
    gemm2_wmma  <<<gT,  B, 0, stream>>>(bufB, W2, bufA, N);
    self_init   <<<gNH, B, 0, stream>>>(bufA, dinv, b2, bufB, N);
    edge_scatter<<<gE4, B, 0, stream>>>(src, dst, dinv, bufA, bufB, E);

    // pool + classifier
    pool_init   <<<(N_GRAPHS * HIDDEN + B - 1) / B, B, 0, stream>>>(sums, cnt);
    pool_scatter<<<gNH, B, 0, stream>>>(bufB, bat, sums, cnt, N);
    finalize    <<<(N_GRAPHS + B - 1) / B, B, 0, stream>>>(sums, cnt, Wl, bl, out);
}